// SQRegularizer_30331059044461
// MI455X (gfx1250) — compile-verified
//
#include <hip/hip_runtime.h>

typedef __attribute__((ext_vector_type(2))) float v2f;
typedef __attribute__((ext_vector_type(8))) float v8f;

#define BB 8
#define PP 16
#define SS 256
#define NN 4096
#define EPS 1e-6f
#define TPB 1024
#define NWAVES 32
#define ITERS 128        // n-tiles per wave (256 tiles / 2 n-chunks)
#define INF_BITS 0x7F800000

static __device__ __forceinline__ int imin32(int a, int b) { return a < b ? a : b; }
static __device__ __forceinline__ int imax32(int a, int b) { return a > b ? a : b; }

// relu entirely in the integer domain: for finite floats, all negatives (incl -0.0)
// view as int < 0, and non-negative floats order as their int bits.
// => max_i32(bits(x), 0) == bits(max(x, 0.0f)), one v_max_i32, no FP canonicalization.
static __device__ __forceinline__ int relu_bits(float x) {
    return imax32((int)__float_as_uint(x), 0);
}

// One workgroup per (b,p); 32 waves: wave w -> s-tile (w & 15), n-chunk (w >> 4).
// Computes:
//   partials[bp*2+0] = sum_n  min_s relu(d2[b,p,s,n]) * assign[b,n,p]
//   partials[bp*2+1] = (1/S) sum_s min_n relu(d2[b,p,s,n])
__global__ __launch_bounds__(TPB) void sqreg_tiles_kernel(
    const float* __restrict__ sp,      // (B,P,S,3)
    const float* __restrict__ pts,     // (B,N,3)
    const float* __restrict__ assign,  // (B,N,P)
    float* __restrict__ partials)      // (B*P, 2)
{
    const int bp     = blockIdx.x;
    const int b      = bp >> 4;
    const int p      = bp & 15;
    const int tid    = threadIdx.x;
    const int wave   = tid >> 5;
    const int lane   = tid & 31;
    const int half   = lane >> 4;     // 0: K=0,1 / rows 0-7 ; 1: K=2,3 / rows 8-15
    const int m      = lane & 15;     // row (A) / col (B,C,D) slot
    const int stile  = wave & 15;     // s-tile owned by this wave
    const int nchunk = wave >> 4;     // n-range half owned by this wave

    __shared__ int colMinI[NN];       // min over s, per n (int view of relu'd f32 bits)
    __shared__ int rowMinI[SS];       // min over n, per s
    __shared__ float red1[NWAVES];
    __shared__ float red2[NWAVES];

    // init mins to +inf bit pattern; all relu'd d2 bits are in [0, INF_BITS]
    for (int n = tid; n < NN; n += TPB) colMinI[n] = INF_BITS;
    if (tid < SS) rowMinI[tid] = INF_BITS;

    // A fragment (16x4 f32): lane m<16 -> {A[m][0],A[m][1]}, lane m+16 -> {A[m][2],A[m][3]}
    //   A = (-2x, -2y, -2z, 1)
    // C accumulator (loop-invariant!): C element (r,lane) is row r+8*half -> a2[row].
    //   D = A*B + C = -2*dot + b2 + a2 = d2, one instruction, zero per-iter C setup.
    v2f afrag;
    v8f cacc;
    {
        const int s = stile * 16 + m;
        const float* q = sp + (((size_t)(b * PP + p) * SS + s) * 3);
        const float ax = q[0], ay = q[1], az = q[2];
        const float a2 = ax * ax + ay * ay + az * az;   // a2 for row m (held by both halves)
        afrag = (half == 0) ? (v2f){-2.f * ax, -2.f * ay} : (v2f){-2.f * az, 1.0f};
        #pragma unroll
        for (int r = 0; r < 8; ++r)
            cacc[r] = __shfl(a2, half * 8 + r, 32);     // broadcast a2 of row r+8*half
    }
    __syncthreads();

    // This wave's point stream: lane column m, n-tiles nchunk*128 .. nchunk*128+127
    const float* __restrict__ pb =
        pts + (size_t)b * NN * 3 + ((size_t)(nchunk * ITERS * 16 + m)) * 3;
    const int colbase = nchunk * (ITERS * 16) + m;

    int rminI[8];
    #pragma unroll
    for (int r = 0; r < 8; ++r) rminI[r] = INF_BITS;

    // software-pipelined point loads (prefetch next tile before current WMMA)
    float x = pb[0], y = pb[1], z = pb[2];

    #pragma unroll 4
    for (int it = 0; it < ITERS; ++it) {
        const int itn = (it < ITERS - 1) ? (it + 1) : it;   // uniform clamped prefetch idx
        const float* qn = pb + (size_t)itn * 48;            // 16 points * 3 floats
        const float nx = qn[0], ny = qn[1], nz = qn[2];

        const float b2 = x * x + y * y + z * z;
        // B fragment (4x16 f32): lane m<16 -> {B[0][m],B[1][m]}, m+16 -> {B[2][m],B[3][m]}
        //   B = (x', y', z', b2)
        const v2f bfrag = (half == 0) ? (v2f){x, y} : (v2f){z, b2};

        v8f d = __builtin_amdgcn_wmma_f32_16x16x4_f32(
            false, afrag, false, bfrag, (short)0, cacc, false, false);

        // relu + all mins in the signed-int domain (v_max_i32 / v_min3_i32 only)
        int t[8];
        #pragma unroll
        for (int r = 0; r < 8; ++r) {
            t[r] = relu_bits(d[r]);
            rminI[r] = imin32(rminI[r], t[r]);
        }
        const int cm =
            imin32(imin32(imin32(t[0], t[1]), imin32(t[2], t[3])),
                   imin32(imin32(t[4], t[5]), imin32(t[6], t[7])));
        // both lane-halves push their 8-row partial min; ds_min_i32 merges (order-free).
        atomicMin(&colMinI[colbase + it * 16], cm);

        x = nx; y = ny; z = nz;
    }

    // reduce row mins across the 16 column slots (xor masks stay within each half)
    #pragma unroll
    for (int mask = 1; mask < 16; mask <<= 1) {
        #pragma unroll
        for (int r = 0; r < 8; ++r)
            rminI[r] = imin32(rminI[r], __shfl_xor(rminI[r], mask, 32));
    }
    if (m == 0) {
        const int sbase = stile * 16 + half * 8;
        #pragma unroll
        for (int r = 0; r < 8; ++r)
            atomicMin(&rowMinI[sbase + r], rminI[r]);   // 2 n-chunk waves merge
    }
    __syncthreads();

    // point->sq partial: sum_n colMin[n] * assign[b,n,p]
    float acc1 = 0.0f;
    for (int n = tid; n < NN; n += TPB)
        acc1 += __uint_as_float((unsigned int)colMinI[n]) *
                assign[((size_t)b * NN + n) * PP + p];
    // sq->point partial: sum_s rowMin[s]
    float acc2 = (tid < SS) ? __uint_as_float((unsigned int)rowMinI[tid]) : 0.0f;

    #pragma unroll
    for (int off = 16; off; off >>= 1) {
        acc1 += __shfl_xor(acc1, off, 32);
        acc2 += __shfl_xor(acc2, off, 32);
    }
    if (lane == 0) { red1[wave] = acc1; red2[wave] = acc2; }
    __syncthreads();
    if (tid == 0) {
        float s1 = 0.f, s2 = 0.f;
        #pragma unroll
        for (int w = 0; w < NWAVES; ++w) { s1 += red1[w]; s2 += red2[w]; }
        partials[bp * 2 + 0] = s1;
        partials[bp * 2 + 1] = s2 * (1.0f / SS);
    }
}

// Final scalar combine: one block, 128 threads (one per (b,p)).
__global__ __launch_bounds__(128) void sqreg_final_kernel(
    const float* __restrict__ partials,  // (128, 2)
    const float* __restrict__ exist,     // (B, P)
    float* __restrict__ out)
{
    const int t = threadIdx.x;           // t = b*16 + p
    float v1  = partials[2 * t + 0];
    float v2e = partials[2 * t + 1] * exist[t];
    float e   = exist[t];

    #pragma unroll
    for (int off = 1; off < 16; off <<= 1) {   // per-b group of 16
        v2e += __shfl_xor(v2e, off, 32);
        e   += __shfl_xor(e,   off, 32);
    }
    #pragma unroll
    for (int off = 16; off; off >>= 1)
        v1 += __shfl_xor(v1, off, 32);

    __shared__ float s1[4];
    __shared__ float sc[8];
    const int wave = t >> 5, lane = t & 31;
    if (lane == 0) s1[wave] = v1;
    if ((lane & 15) == 0) sc[t >> 4] = v2e / fmaxf(e, EPS);
    __syncthreads();

    if (t == 0) {
        float p2s = 0.f;
        #pragma unroll
        for (int w = 0; w < 4; ++w) p2s += s1[w];
        float s2p = 0.f;
        #pragma unroll
        for (int bb = 0; bb < 8; ++bb) s2p += sc[bb];
        out[0] = p2s * (1.0f / (BB * (float)NN)) + s2p * (1.0f / BB);
    }
}

extern "C" void kernel_launch(void* const* d_in, const int* in_sizes, int n_in,
                              void* d_out, int out_size, void* d_ws, size_t ws_size,
                              hipStream_t stream) {
    const float* sp     = (const float*)d_in[0];  // surface_points (8,16,256,3)
    const float* pts    = (const float*)d_in[1];  // points         (8,4096,3)
    const float* assign = (const float*)d_in[2];  // assign_matrix  (8,4096,16)
    const float* exist  = (const float*)d_in[3];  // exist          (8,16)
    float* out = (float*)d_out;
    float* partials = (float*)d_ws;               // 128*2 floats

    sqreg_tiles_kernel<<<BB * PP, TPB, 0, stream>>>(sp, pts, assign, partials);
    sqreg_final_kernel<<<1, 128, 0, stream>>>(partials, exist, out);
}